// ReaxFF_nn_force_39986145526132
// MI455X (gfx1250) — compile-verified
//
#include <hip/hip_runtime.h>
#include <stdint.h>

// Problem geometry (fixed by the reference).
#define ATOMS 2048
#define CH    512            // j-chunk staged per TDM round
#define NCH   (ATOMS / CH)   // 4 chunks per row
#define TPB   256            // 8 wave32 per block
#define NPAR  9              // rosi,bo1,bo2,ropi,bo3,bo4,ropp,bo5,bo6
#define JPT   (CH / TPB)     // 2 j's per thread per chunk

typedef unsigned int v4u_t __attribute__((ext_vector_type(4)));
typedef int          v8i_t __attribute__((ext_vector_type(8)));
typedef int          v4i_t __attribute__((ext_vector_type(4)));
typedef float        v2f_t __attribute__((ext_vector_type(2)));
typedef float        v8f_t __attribute__((ext_vector_type(8)));

// ---------------------------------------------------------------------------
// Tensor Data Mover: async copy of CH contiguous f32 from global -> LDS.
// D# packing per CDNA5 ISA §8.3 (group0) / §8.4 (group1):
//   g0: [1:0]=count=1, [63:32]=lds_addr, [120:64]=global_addr, [127:126]=2
//   g1: data_size=2 (4B), tensor_dim0=CH, tensor_dim1=1, tile_dim0=CH,
//       tile_dim1/2=0 (unused), tensor_dim0_stride=CH, no pad/iterate/mcast
// ---------------------------------------------------------------------------
__device__ __forceinline__ void tdm_row_chunk_to_lds(uint32_t lds_addr,
                                                     const float* gsrc)
{
    uint64_t ga = (uint64_t)(uintptr_t)gsrc;
    v4u_t g0;
    g0.x = 1u;                                                   // count=1, user desc
    g0.y = lds_addr;                                             // lds_addr bytes
    g0.z = (uint32_t)ga;                                         // global_addr[31:0]
    g0.w = (uint32_t)((ga >> 32) & 0x01FFFFFFull) | (2u << 30);  // [56:32] | type=2
    v8i_t g1;
    g1[0] = 2 << 16;            // workgroup_mask=0, data_size=4B, no barrier/pad
    g1[1] = (int)(CH << 16);    // atomic_barrier_addr=0 | tensor_dim0.lo=CH
    g1[2] = 1 << 16;            // tensor_dim0.hi=0 | tensor_dim1.lo=1
    g1[3] = (int)(CH << 16);    // tensor_dim1.hi=0 | tile_dim0=CH
    g1[4] = 0;                  // tile_dim1=0 (unused) | tile_dim2=0
    g1[5] = CH;                 // tensor_dim0_stride.lo
    g1[6] = 0;                  // stride0.hi | stride1.lo
    g1[7] = 0;                  // stride1.hi
    v4i_t gz = {0, 0, 0, 0};    // groups 2/3 unused (<=2D tensor)
#if __has_include(<hip/amd_detail/amd_gfx1250_TDM.h>)
    // amdgpu-toolchain (clang-23 / therock-10.0): 6-arg builtin
    v8i_t gz8 = {0, 0, 0, 0, 0, 0, 0, 0};
    __builtin_amdgcn_tensor_load_to_lds(g0, g1, gz, gz, gz8, 0);
#else
    // ROCm 7.2 (clang-22): 5-arg builtin
    __builtin_amdgcn_tensor_load_to_lds(g0, g1, gz, gz, 0);
#endif
}

// taper(e, BOTOL, 2*BOTOL), faithful branchless translation.
__device__ __forceinline__ float taper2x(float e)
{
    const float rmin = 0.002f, rmax = 0.004f;
    const float rterm = -1.25e8f;                 // 1/(rmin-rmax)^3 exactly
    float r3  = (e > rmax) ? 1.0f : 0.0f;
    bool  ok  = (e <= rmax) && (e > rmin);
    float r2  = ok ? e : 0.0f;
    float r20 = ok ? 1.0f : 0.0f;
    float rm  = rmin * r20;
    float rd  = rm - r2;
    float trm1 = rm + 2.0f * r2 - 3.0f * rmax * r20;
    return rterm * rd * rd * trm1 + r3;
}

extern "C" __global__ __launch_bounds__(TPB)
void reaxff_bop_rows(const float* __restrict__ x,
                     const float* __restrict__ cell,
                     const float* __restrict__ rcell,
                     const float* __restrict__ rosi, const float* __restrict__ bo1,
                     const float* __restrict__ bo2,  const float* __restrict__ ropi,
                     const float* __restrict__ bo3,  const float* __restrict__ bo4,
                     const float* __restrict__ ropp, const float* __restrict__ bo5,
                     const float* __restrict__ bo6,
                     float* __restrict__ out)
{
    __shared__ __align__(16) float sP[2][NPAR][CH];   // 36 KB double buffer
    __shared__ float sRed[TPB / 32][3];

    const int i   = blockIdx.x;     // row (atom i)
    const int tid = threadIdx.x;

    // Tiny uniform data: cell/rcell (9 f32 each) and x_i — L2/scalar-cache hits.
    float rc[9], cl[9];
#pragma unroll
    for (int k = 0; k < 9; ++k) { rc[k] = rcell[k]; cl[k] = cell[k]; }
    const float xi0 = x[3 * i + 0], xi1 = x[3 * i + 1], xi2 = x[3 * i + 2];

    const float* par[NPAR] = { rosi, bo1, bo2, ropi, bo3, bo4, ropp, bo5, bo6 };
    const uint32_t lds0 = (uint32_t)(uintptr_t)(&sP[0][0][0]); // low32 of flat = LDS off
    const size_t rowoff = (size_t)i * ATOMS;

    // Prologue: kick chunk 0 into buffer 0 (TDM ignores EXEC; wave0 issues).
    if (tid == 0) {
#pragma unroll
        for (int p = 0; p < NPAR; ++p)
            tdm_row_chunk_to_lds(lds0 + (uint32_t)(p * CH * 4), par[p] + rowoff);
    }

    float asi = 0.0f, api = 0.0f, app = 0.0f;

    for (int ck = 0; ck < NCH; ++ck) {
        const int buf = ck & 1;
        if (tid == 0) {
            if (ck + 1 < NCH) {
                const int nb = (ck + 1) & 1;
#pragma unroll
                for (int p = 0; p < NPAR; ++p)
                    tdm_row_chunk_to_lds(lds0 + (uint32_t)((nb * NPAR + p) * CH * 4),
                                         par[p] + rowoff + (size_t)(ck + 1) * CH);
                // 9 newer ops in flight; TENSORcnt is in-order per wave, so
                // waiting to <=9 guarantees chunk ck's 9 loads have landed.
                __builtin_amdgcn_s_wait_tensorcnt((short)NPAR);
            } else {
                __builtin_amdgcn_s_wait_tensorcnt((short)0);
            }
        }
        __syncthreads();   // publish TDM-written LDS tile to all 8 waves

        const int l0 = tid * JPT;  // 2 consecutive j's -> ds_load_b64 per param
        const float2 v_rosi = *(const float2*)&sP[buf][0][l0];
        const float2 v_bo1  = *(const float2*)&sP[buf][1][l0];
        const float2 v_bo2  = *(const float2*)&sP[buf][2][l0];
        const float2 v_ropi = *(const float2*)&sP[buf][3][l0];
        const float2 v_bo3  = *(const float2*)&sP[buf][4][l0];
        const float2 v_bo4  = *(const float2*)&sP[buf][5][l0];
        const float2 v_ropp = *(const float2*)&sP[buf][6][l0];
        const float2 v_bo5  = *(const float2*)&sP[buf][7][l0];
        const float2 v_bo6  = *(const float2*)&sP[buf][8][l0];
        const int jbase = ck * CH + l0;

#pragma unroll
        for (int u = 0; u < JPT; ++u) {
            const int j = jbase + u;
            const float p_rosi = u ? v_rosi.y : v_rosi.x;
            const float p_bo1  = u ? v_bo1.y  : v_bo1.x;
            const float p_bo2  = u ? v_bo2.y  : v_bo2.x;
            const float p_ropi = u ? v_ropi.y : v_ropi.x;
            const float p_bo3  = u ? v_bo3.y  : v_bo3.x;
            const float p_bo4  = u ? v_bo4.y  : v_bo4.x;
            const float p_ropp = u ? v_ropp.y : v_ropp.x;
            const float p_bo5  = u ? v_bo5.y  : v_bo5.x;
            const float p_bo6  = u ? v_bo6.y  : v_bo6.x;

            // min-image distance through general cell/rcell, as in reference
            float dx0 = xi0 - x[3 * j + 0];
            float dx1 = xi1 - x[3 * j + 1];
            float dx2 = xi2 - x[3 * j + 2];
            float f0 = dx0 * rc[0] + dx1 * rc[3] + dx2 * rc[6];
            float f1 = dx0 * rc[1] + dx1 * rc[4] + dx2 * rc[7];
            float f2 = dx0 * rc[2] + dx1 * rc[5] + dx2 * rc[8];
            f0 = (f0 - 0.5f > 0.0f) ? f0 - 1.0f : f0;
            f0 = (f0 + 0.5f < 0.0f) ? f0 + 1.0f : f0;
            f1 = (f1 - 0.5f > 0.0f) ? f1 - 1.0f : f1;
            f1 = (f1 + 0.5f < 0.0f) ? f1 + 1.0f : f1;
            f2 = (f2 - 0.5f > 0.0f) ? f2 - 1.0f : f2;
            f2 = (f2 + 0.5f < 0.0f) ? f2 + 1.0f : f2;
            float w0 = f0 * cl[0] + f1 * cl[3] + f2 * cl[6];
            float w1 = f0 * cl[1] + f1 * cl[4] + f2 * cl[7];
            float w2 = f0 * cl[2] + f1 * cl[5] + f2 * cl[8];
            float r2d = w0 * w0 + w1 * w1 + w2 * w2;
            float r   = __builtin_sqrtf(r2d + 1e-9f);
            float frcv = (r2d < 25.0f) ? 1.0f : 0.0f;   // r<RCUT recomputed

            // bond-order chain: exp(b * (r/ro)^b2) via v_log/v_exp
            float e1 = 1.002f * __expf(p_bo1 * __powf(__fdividef(r, p_rosi), p_bo2));
            float e2 = __expf(p_bo3 * __powf(__fdividef(r, p_ropi), p_bo4)) * frcv;
            float e3 = __expf(p_bo5 * __powf(__fdividef(r, p_ropp), p_bo6)) * frcv;

            float fsi = taper2x(e1) * (e1 - 0.002f);
            float fpi = taper2x(e2) * e2;
            float fpp = taper2x(e3) * e3;
            float ey  = (j == i) ? 0.0f : 1.0f;          // eye recomputed
            asi += fsi * ey;
            api += fpi * ey;
            app += fpp * ey;
        }
        __syncthreads();   // buffer `buf` free for the chunk ck+2 TDM round
    }

    // ------------------------------------------------------------------
    // Wave-level reduction via V_WMMA_F32_16X16X4_F32 (exact f32 math).
    // A (16x4) = 2 VGPRs/wave: A[m][0]=asi(m), A[m][1]=api(m),
    //                          A[m][2]=asi(m+16), A[m][3]=api(m+16).
    // B (4x16) selector: col0 sums K in {0,2}, col1 sums K in {1,3}:
    //   B vgpr0 (K=0 lanes0-15, K=2 lanes16-31) = 1 at lanes {0,16}
    //   B vgpr1 (K=1, K=3)                      = 1 at lanes {1,17}
    // => D[m][0]=asi(m)+asi(m+16), D[m][1]=api(m)+api(m+16).
    // Summing the 8 D VGPRs per lane collapses M: lane0 holds Sum(m=0..7),
    // lane16 holds Sum(m=8..15) (col 0 = asi; col 1 / lanes 1,17 = api).
    // EXEC is all ones here (uniform control flow), as WMMA requires.
    // ------------------------------------------------------------------
    const int ln = tid & 31;
    v2f_t A1; A1.x = asi; A1.y = api;
    v2f_t A2; A2.x = app; A2.y = 0.0f;
    v2f_t B;
    B.x = (ln == 0 || ln == 16) ? 1.0f : 0.0f;
    B.y = (ln == 1 || ln == 17) ? 1.0f : 0.0f;
    v8f_t Zc = {0.0f, 0.0f, 0.0f, 0.0f, 0.0f, 0.0f, 0.0f, 0.0f};
    v8f_t D1 = __builtin_amdgcn_wmma_f32_16x16x4_f32(false, A1, false, B,
                                                     (short)0, Zc, false, false);
    v8f_t D2 = __builtin_amdgcn_wmma_f32_16x16x4_f32(false, A2, false, B,
                                                     (short)0, Zc, false, false);
    float t1 = ((D1[0] + D1[1]) + (D1[2] + D1[3])) +
               ((D1[4] + D1[5]) + (D1[6] + D1[7]));
    float t2 = ((D2[0] + D2[1]) + (D2[2] + D2[3])) +
               ((D2[4] + D2[5]) + (D2[6] + D2[7]));
    const float wsi = __shfl(t1, 0, 32) + __shfl(t1, 16, 32);
    const float wpi = __shfl(t1, 1, 32) + __shfl(t1, 17, 32);
    const float wpp = __shfl(t2, 0, 32) + __shfl(t2, 16, 32);

    const int wv = tid >> 5;
    if (ln == 0) { sRed[wv][0] = wsi; sRed[wv][1] = wpi; sRed[wv][2] = wpp; }
    __syncthreads();
    if (tid == 0) {
        float s0 = 0.0f, s1 = 0.0f, s2 = 0.0f;
#pragma unroll
        for (int w = 0; w < TPB / 32; ++w) {
            s0 += sRed[w][0]; s1 += sRed[w][1]; s2 += sRed[w][2];
        }
        out[0 * ATOMS + i] = s0 + s1 + s2;   // Deltap
        out[1 * ATOMS + i] = s0;             // D_si
        out[2 * ATOMS + i] = s1;             // D_pi
        out[3 * ATOMS + i] = s2;             // D_pp
    }
}

extern "C" void kernel_launch(void* const* d_in, const int* in_sizes, int n_in,
                              void* d_out, int out_size, void* d_ws, size_t ws_size,
                              hipStream_t stream)
{
    (void)in_sizes; (void)n_in; (void)out_size; (void)d_ws; (void)ws_size;
    const float* x     = (const float*)d_in[0];
    const float* cell  = (const float*)d_in[1];
    const float* rcell = (const float*)d_in[2];
    const float* rosi  = (const float*)d_in[3];
    const float* bo1   = (const float*)d_in[4];
    const float* bo2   = (const float*)d_in[5];
    const float* ropi  = (const float*)d_in[6];
    const float* bo3   = (const float*)d_in[7];
    const float* bo4   = (const float*)d_in[8];
    const float* ropp  = (const float*)d_in[9];
    const float* bo5   = (const float*)d_in[10];
    const float* bo6   = (const float*)d_in[11];
    // d_in[12] (frc) and d_in[13] (eye) are recomputed on the fly to save
    // ~33 MB of HBM traffic per call on this bandwidth-bound kernel.
    float* out = (float*)d_out;

    reaxff_bop_rows<<<ATOMS, TPB, 0, stream>>>(x, cell, rcell,
                                               rosi, bo1, bo2,
                                               ropi, bo3, bo4,
                                               ropp, bo5, bo6,
                                               out);
}